// PDFSampler_88837103551055
// MI455X (gfx1250) — compile-verified
//
#include <hip/hip_runtime.h>
#include <stdint.h>

// ---------------- problem constants (from reference) ----------------
#define N_RAYS      262144
#define N_COARSE    64
#define NUM_SAMPLES 128
#define NUM_BINS    (NUM_SAMPLES + 1)          // 129 u-samples
#define N_EXIST     (N_COARSE + 1)             // 65 existing bin edges / cdf entries
#define OUT_PER_RAY (N_EXIST + NUM_BINS - 1)   // 193 sorted outputs per ray
#define HIST_PAD    0.01f
#define EPS_PAD     1e-5f

#define WAVES_PER_BLOCK 8
#define THREADS        (WAVES_PER_BLOCK * 32)
#define LDS_STRIDE      468   // cdf[65] @0, exist[65] @72, newb[129] @140, stage[193] @272

// ---------------- CDNA5 async global<->LDS paths ----------------
#if defined(__has_builtin)
#  if __has_builtin(__builtin_amdgcn_global_load_async_to_lds_b64) && \
      __has_builtin(__builtin_amdgcn_global_load_async_to_lds_b32)
#    define HAVE_ASYNC_LD 1
#  endif
#  if __has_builtin(__builtin_amdgcn_global_store_async_from_lds_b32)
#    define HAVE_ASYNC_ST 1
#  endif
#endif

// Builtin param types (from clang diagnostics): bN transfers are typed —
// b32 -> int*, b64 -> pointer to 2 x i32 GCC vector. Global arg AS1, LDS arg AS3.
typedef int v2i __attribute__((vector_size(2 * sizeof(int))));
typedef __attribute__((address_space(1))) v2i* g_v2i_p;
typedef __attribute__((address_space(3))) v2i* l_v2i_p;
typedef __attribute__((address_space(1))) int* g_i32_p;
typedef __attribute__((address_space(3))) int* l_i32_p;

// Flat addresses of LDS objects carry the LDS byte offset in the low 32 bits
// (ISA 10.2 aperture mapping); integer round-trip is a
// compile-safe addrspace cast.
static __device__ __forceinline__ g_v2i_p as_g2(const void* p) {
  return (g_v2i_p)(uintptr_t)p;
}
static __device__ __forceinline__ l_v2i_p as_l2(const void* p) {
  return (l_v2i_p)(uint32_t)(uintptr_t)p;
}
static __device__ __forceinline__ g_i32_p as_g1(const void* p) {
  return (g_i32_p)(uintptr_t)p;
}
static __device__ __forceinline__ l_i32_p as_l1(const void* p) {
  return (l_i32_p)(uint32_t)(uintptr_t)p;
}

__global__ __launch_bounds__(THREADS)
void PDFSampler_kernel(const float* __restrict__ weights,
                       const float* __restrict__ bins,
                       const float* __restrict__ max_bin,
                       float* __restrict__ out) {
  __shared__ float smem[WAVES_PER_BLOCK * LDS_STRIDE];

  const int lane = threadIdx.x & 31;
  const int wave = threadIdx.x >> 5;
  const int ray  = blockIdx.x * WAVES_PER_BLOCK + wave;   // one wave32 per ray

  float* cdf   = &smem[wave * LDS_STRIDE];  // [0..64]
  float* exist = cdf + 72;                  // [0..64] existing bin edges
  float* newb  = cdf + 140;                 // [0..128] sampled bins
  float* stage = cdf + 272;                 // [0..192] merged, rank-ordered

  const float* wrow = weights + (size_t)ray * N_COARSE;
  const float* brow = bins    + (size_t)ray * N_COARSE;

  // ---- stage existing_bins into this wave's LDS slice ----
#if defined(HAVE_ASYNC_LD)
  // 32 lanes x 8B = the full 256B bin row, DMA'd while we do the scan below.
  __builtin_amdgcn_global_load_async_to_lds_b64(
      as_g2(brow + 2 * lane), as_l2(exist + 2 * lane), 0, 0);
  if (lane == 0) {
    __builtin_amdgcn_global_load_async_to_lds_b32(
        as_g1(max_bin + ray), as_l1(exist + N_COARSE), 0, 0);
  }
#else
  {
    float2 bv = *(const float2*)(brow + 2 * lane);
    exist[2 * lane]     = bv.x;
    exist[2 * lane + 1] = bv.y;
    if (lane == 0) exist[N_COARSE] = max_bin[ray];
  }
#endif

  // ---- weights -> padded PDF -> CDF via wave32 inclusive scan ----
  float2 wv = *(const float2*)(wrow + 2 * lane);   // coalesced global_load_b64
  float a = wv.x + HIST_PAD;
  float b = wv.y + HIST_PAD;
  float local = a + b;
  float incl  = local;
#pragma unroll
  for (int off = 1; off < 32; off <<= 1) {
    float t = __shfl_up(incl, off, 32);
    if (lane >= off) incl += t;
  }
  float total = __shfl(incl, 31, 32);             // w_sum (pre-extra-padding)
  float pad   = fmaxf(EPS_PAD - total, 0.0f);
  float inv   = 1.0f / (total + pad);
  float padc  = pad * (1.0f / (float)N_COARSE);
  float excl  = incl - local;

  int k1 = 2 * lane + 1, k2 = 2 * lane + 2;
  cdf[k1] = fminf((excl + a + (float)k1 * padc) * inv, 1.0f);
  cdf[k2] = fminf((incl     + (float)k2 * padc) * inv, 1.0f);
  if (lane == 0) cdf[0] = 0.0f;

#if defined(HAVE_ASYNC_LD)
#  if defined(__has_builtin) && __has_builtin(__builtin_amdgcn_s_wait_asynccnt)
  __builtin_amdgcn_s_wait_asynccnt(0);
  asm volatile("" ::: "memory");
#  else
  asm volatile("s_wait_asynccnt 0" ::: "memory");
#  endif
#endif
  // Each wave touches only its own LDS slice and intra-wave DS ops are
  // in-order on CDNA5 -> no workgroup barrier needed anywhere.

  // ---- inverse-CDF sampling: searchsorted(cdf, u, side='right') ----
  for (int j = lane; j < NUM_BINS; j += 32) {
    float u = ((float)j + 0.5f) * (1.0f / (float)NUM_BINS);
    int lo = 0, hi = N_EXIST;                     // idx in [1, 65] since cdf[0]=0<u
    while (lo < hi) {
      int mid = (lo + hi) >> 1;
      if (cdf[mid] <= u) lo = mid + 1; else hi = mid;
    }
    int below = lo - 1;  if (below > N_COARSE) below = N_COARSE;
    int above = lo;      if (above > N_COARSE) above = N_COARSE;
    float c0 = cdf[below],  c1 = cdf[above];
    float e0 = exist[below], e1 = exist[above];
    float den = c1 - c0;
    float ts  = (den > 0.0f) ? (u - c0) / den : 0.0f;
    ts = fminf(fmaxf(ts, 0.0f), 1.0f);
    newb[j] = e0 + ts * (e1 - e0);
  }

  // ---- merge two sorted arrays by rank into LDS stage; rank 193 dropped ----
  // Ranks 0..193 form a permutation of the 194 elements (ties: existing first),
  // so stage[0..192] is fully written.
  for (int i = lane; i < N_EXIST; i += 32) {       // existing: ties go first
    float x = exist[i];
    int lo = 0, hi = NUM_BINS;
    while (lo < hi) {
      int mid = (lo + hi) >> 1;
      if (newb[mid] < x) lo = mid + 1; else hi = mid;
    }
    int rank = i + lo;
    if (rank < OUT_PER_RAY) stage[rank] = x;
  }
  for (int j = lane; j < NUM_BINS; j += 32) {      // new samples: ties go after
    float x = newb[j];
    int lo = 0, hi = N_EXIST;
    while (lo < hi) {
      int mid = (lo + hi) >> 1;
      if (exist[mid] <= x) lo = mid + 1; else hi = mid;
    }
    int rank = j + lo;
    if (rank < OUT_PER_RAY) stage[rank] = x;
  }

  // ---- drain stage -> global, fully coalesced ----
  float* orow = out + (size_t)ray * OUT_PER_RAY;   // only 4B-aligned (193 floats/row)
#if defined(HAVE_ASYNC_ST)
  // Make the wave's DS rank-writes visible to the async DMA engine before it
  // reads LDS (compiler can't see this cross-engine dependency).
  asm volatile("s_wait_dscnt 0" ::: "memory");
#pragma unroll
  for (int r = 0; r < 7; ++r) {                    // 7 x 32 lanes covers 193 (+partial)
    int j = r * 32 + lane;
    if (j < OUT_PER_RAY) {
      __builtin_amdgcn_global_store_async_from_lds_b32(
          as_g1(orow + j), as_l1(stage + j), 0, 0);
    }
  }
  // s_endpgm performs an implicit wait-idle, covering outstanding ASYNCcnt.
#else
  for (int j = lane; j < OUT_PER_RAY; j += 32) {
    orow[j] = stage[j];                            // coalesced 128B bursts
  }
#endif
}

extern "C" void kernel_launch(void* const* d_in, const int* in_sizes, int n_in,
                              void* d_out, int out_size, void* d_ws, size_t ws_size,
                              hipStream_t stream) {
  (void)in_sizes; (void)n_in; (void)out_size; (void)d_ws; (void)ws_size;
  const float* weights = (const float*)d_in[0];  // (N_RAYS, 64, 1)
  const float* bins    = (const float*)d_in[1];  // (N_RAYS, 64, 1)
  const float* max_bin = (const float*)d_in[2];  // (N_RAYS, 1)
  float* out = (float*)d_out;                    // (N_RAYS, 193)

  dim3 grid(N_RAYS / WAVES_PER_BLOCK);
  PDFSampler_kernel<<<grid, THREADS, 0, stream>>>(weights, bins, max_bin, out);
}